// LGConv_89103391523471
// MI455X (gfx1250) — compile-verified
//
#include <hip/hip_runtime.h>
#include <hip/hip_bf16.h>

// ---------------------------------------------------------------------------
// LGConv (SGC-style): deg/norm -> 2x (pre-norm gather, scatter-add, post-norm)
// -> S = a0*f0 + a1*f1 + a2*f2 -> out = S @ W^T + 3*b  (WMMA f32 16x16x4)
// ---------------------------------------------------------------------------

#define LG_N     100000
#define LG_E     1600000
#define LG_D     32
#define LG_K     2

typedef __attribute__((ext_vector_type(2))) float v2f;
typedef __attribute__((ext_vector_type(8))) float v8f;

static __device__ __forceinline__ void atomic_add_f32(float* p, float v) {
  __hip_atomic_fetch_add(p, v, __ATOMIC_RELAXED, __HIP_MEMORY_SCOPE_AGENT);
}

// Zero a float region (float4-vectorized; count4 = #float4s).
__global__ void lg_zero4(float4* __restrict__ p, int count4) {
  int i = blockIdx.x * blockDim.x + threadIdx.x;
  if (i < count4) p[i] = make_float4(0.f, 0.f, 0.f, 0.f);
}

// S = alpha[0] * feat  (float4-vectorized over N*32 elements)
__global__ void lg_init_S(const float4* __restrict__ feat, float4* __restrict__ S,
                          const float* __restrict__ alpha, int count4) {
  int i = blockIdx.x * blockDim.x + threadIdx.x;
  if (i >= count4) return;
  float a0 = alpha[0];
  float4 f = feat[i];
  S[i] = make_float4(a0 * f.x, a0 * f.y, a0 * f.z, a0 * f.w);
}

// In-degree via integer atomics into the norm buffer (viewed as uint).
__global__ void lg_degree(const int* __restrict__ dst, unsigned* __restrict__ deg, int e) {
  int i = blockIdx.x * blockDim.x + threadIdx.x;
  if (i < e) atomicAdd(&deg[dst[i]], 1u);
}

// norm[i] = 1/sqrt(max(deg,1)) in place (uint -> float reuse of buffer).
__global__ void lg_norm(float* __restrict__ norm, int n) {
  int i = blockIdx.x * blockDim.x + threadIdx.x;
  if (i >= n) return;
  unsigned d = ((const unsigned*)norm)[i];
  float df = (d > 0u) ? (float)d : 1.0f;
  norm[i] = 1.0f / sqrtf(df);
}

// SpMM scatter: hout[dst] += norm[src] * hin[src].  8 threads/edge, float4 each.
// A wave32 covers 4 edges x 8 chunks: gathers are 4x128B segments, and the 8
// atomics per edge hit consecutive addresses of one dst row (L2-friendly).
__global__ void lg_scatter(const float* __restrict__ hin, const float* __restrict__ norm,
                           const int* __restrict__ src, const int* __restrict__ dst,
                           float* __restrict__ hout, int e_total) {
  int t = blockIdx.x * blockDim.x + threadIdx.x;
  int e = t >> 3;
  if (e >= e_total) return;
  int c = (t & 7) << 2;
  int s = src[e];
  int d = dst[e];
  float ns = norm[s];
  float4 v = *(const float4*)(hin + (size_t)s * LG_D + c);
  float* o = hout + (size_t)d * LG_D + c;
  atomic_add_f32(o + 0, ns * v.x);
  atomic_add_f32(o + 1, ns * v.y);
  atomic_add_f32(o + 2, ns * v.z);
  atomic_add_f32(o + 3, ns * v.w);
}

// Post-normalize at dst and accumulate into S:  S += alpha_k * norm[row] * hraw.
// Optionally writes the normalized hop features back (needed as next hop input).
__global__ void lg_post_acc(float* __restrict__ hraw, const float* __restrict__ norm,
                            float* __restrict__ S, const float* __restrict__ alpha,
                            int alpha_idx, int write_h, int count4) {
  int i = blockIdx.x * blockDim.x + threadIdx.x;   // one float4 per thread
  if (i >= count4) return;
  int row = i >> 3;                                // 8 float4s per 32-wide row
  float nr = norm[row];
  float ak = alpha[alpha_idx];
  float4 h = ((const float4*)hraw)[i];
  float4 hn = make_float4(nr * h.x, nr * h.y, nr * h.z, nr * h.w);
  if (write_h) ((float4*)hraw)[i] = hn;
  float4 s = ((const float4*)S)[i];
  s.x += ak * hn.x; s.y += ak * hn.y; s.z += ak * hn.z; s.w += ak * hn.w;
  ((float4*)S)[i] = s;
}

// out[N,32] = S[N,32] @ W^T[32,32] + 3*b   via V_WMMA_F32_16X16X4_F32.
// Block = 128 threads = 4 waves; each wave owns one 16x16 output tile:
//   wave>>1 selects the 16-row sub-tile, wave&1 selects output cols 0-15/16-31.
// N % 32 == 0 -> tiling is exact, so the epilogue stores are unguarded:
// one row base address + 8 global_store_b32 with immediate offsets.
__global__ void lg_wmma_out(const float* __restrict__ S, const float* __restrict__ W,
                            const float* __restrict__ bias, float* __restrict__ out,
                            int n) {
  int wave = threadIdx.x >> 5;
  int lane = threadIdx.x & 31;
  int m0 = (blockIdx.x * 2 + (wave >> 1)) * 16;   // row tile base
  int n0 = (wave & 1) * 16;                        // col tile base
  int row = lane & 15;
  int khi = (lane >> 4) << 1;                      // lanes 0-15 -> K+0, 16-31 -> K+2
  int col = n0 + (lane & 15);

  int gr = m0 + row;
  if (gr >= n) gr = n - 1;                         // clamp (keeps EXEC all-ones)
  const float* srow = S + (size_t)gr * LG_D;
  const float* wrow = W + (size_t)col * LG_D;      // W row 'col' == column of W^T

  v8f c = {};
#pragma unroll
  for (int k0 = 0; k0 < LG_D; k0 += 4) {
    int kb = k0 + khi;
    v2f a = *(const v2f*)(srow + kb);              // {S[gr,kb], S[gr,kb+1]}
    v2f b = *(const v2f*)(wrow + kb);              // {W^T[kb,col], W^T[kb+1,col]}
    c = __builtin_amdgcn_wmma_f32_16x16x4_f32(false, a, false, b,
                                              (short)0, c, false, false);
  }

  float bv = (float)(LG_K + 1) * bias[col];        // bias added once per hop term
  int mbase = m0 + ((lane >> 4) ? 8 : 0);          // D: VGPR r -> M = r (+8 hi half)
  float* orow = out + (size_t)mbase * LG_D + col;  // stride LG_D between D rows
#pragma unroll
  for (int r = 0; r < 8; ++r) {
    orow[(size_t)r * LG_D] = c[r] + bv;            // immediate-offset stores
  }
}

extern "C" void kernel_launch(void* const* d_in, const int* in_sizes, int n_in,
                              void* d_out, int out_size, void* d_ws, size_t ws_size,
                              hipStream_t stream) {
  const float* feat  = (const float*)d_in[0];   // [N,32]
  const int*   src   = (const int*)d_in[1];     // [E]
  const int*   dst   = (const int*)d_in[2];     // [E]
  const float* W     = (const float*)d_in[3];   // [32,32]
  const float* bias  = (const float*)d_in[4];   // [32]
  const float* alpha = (const float*)d_in[5];   // [K+1]
  float*       out   = (float*)d_out;

  const int n = LG_N, e = LG_E;

  // Workspace layout (floats): [norm n][h1 32n][h2 32n][S 32n]
  float* ws   = (float*)d_ws;
  float* norm = ws;
  float* h1   = ws + n;
  float* h2   = ws + n + (size_t)32 * n;
  float* S    = ws + n + (size_t)64 * n;

  const int B = 256;
  const int nd4  = (n * LG_D) / 4;     // float4s per feature matrix
  const int z4   = (n + 64 * n) / 4;   // zero norm+h1+h2 (n%4==0)

  // 1) zero norm (deg) + h1 + h2; init S = alpha0 * feat
  lg_zero4<<<(z4 + B - 1) / B, B, 0, stream>>>((float4*)ws, z4);
  lg_init_S<<<(nd4 + B - 1) / B, B, 0, stream>>>((const float4*)feat, (float4*)S,
                                                 alpha, nd4);
  // 2) degrees + norm
  lg_degree<<<(e + B - 1) / B, B, 0, stream>>>(dst, (unsigned*)norm, e);
  lg_norm<<<(n + B - 1) / B, B, 0, stream>>>(norm, n);

  // 3) hop 1: h1_raw = scatter(norm[src]*feat[src]); h1 = norm*h1_raw; S += a1*h1
  {
    int t = e * 8;
    lg_scatter<<<(t + B - 1) / B, B, 0, stream>>>(feat, norm, src, dst, h1, e);
    lg_post_acc<<<(nd4 + B - 1) / B, B, 0, stream>>>(h1, norm, S, alpha, 1, 1, nd4);
  }
  // 4) hop 2: h2_raw = scatter(norm[src]*h1[src]);  S += a2 * norm * h2_raw
  {
    int t = e * 8;
    lg_scatter<<<(t + B - 1) / B, B, 0, stream>>>(h1, norm, src, dst, h2, e);
    lg_post_acc<<<(nd4 + B - 1) / B, B, 0, stream>>>(h2, norm, S, alpha, 2, 0, nd4);
  }

  // 5) out = S @ W^T + 3*b via f32 WMMA (N % 32 == 0 -> exact tiling)
  {
    int blocks = (n + 31) / 32;        // 2 row-tiles of 16 per block
    lg_wmma_out<<<blocks, 128, 0, stream>>>(S, W, bias, out, n);
  }
}